// customConvMP_61383672594959
// MI455X (gfx1250) — compile-verified
//
#include <hip/hip_runtime.h>
#include <stdint.h>

// ---------------------------------------------------------------------------
// spikeMP "conv": B=8, H=W=32, C=3, k=3 -> D=27, N=2D=54, F=64.
// Per (pixel, f): sort 54 values, cumsum threshold, twice, relu of difference.
// Pure VALU sort-network workload (no matmul -> WMMA inapplicable); CDNA5
// async-tensor path stages the weight matrix into LDS
// (global_load_async_to_lds_b128 + s_wait_asynccnt).
//
// Sort: Batcher merge-exchange network for exactly n=54 (441 comparators).
// The comparator list is built by a constexpr generator and applied through
// template recursion, so every array index is a compile-time constant by
// construction -> SROA must keep the 54-slot array in VGPRs (rounds 2/3
// showed the nested-loop form partially failing to unroll, demoting the
// array to scratch: ~873 vmem spill ops).
// ---------------------------------------------------------------------------

#define BIGV 999.0f
#define NZ   54

// ---------------- comparator table (Knuth 5.2.2 Algorithm M, n=54) ----------
struct CEPair { unsigned char a, b; };
struct CmpTable { CEPair c[512]; int n; };

constexpr CmpTable make_table() {
    CmpTable t{};
    int idx = 0;
    for (int k = 5; k >= 0; --k) {                 // p = 2^k
        const int p = 1 << k;
        for (int j = 0; j <= 5 - k; ++j) {
            const int d = (j == 0) ? p : ((1 << (6 - j)) - p);
            const int r = (j == 0) ? 0 : p;
            for (int i = 0; i + d < NZ; ++i) {
                if ((i & p) == r) {
                    t.c[idx].a = (unsigned char)i;
                    t.c[idx].b = (unsigned char)(i + d);
                    ++idx;
                }
            }
        }
    }
    t.n = idx;
    return t;
}

constexpr CmpTable TBL = make_table();
constexpr int NCMP = TBL.n;                        // 441 for n=54
static_assert(NCMP > 0 && NCMP <= 512, "comparator table overflow");

// Template-recursive network: indices are constexpr by construction.
template <int T>
struct SortNet {
    static __device__ __forceinline__ void run(float (&s)[NZ]) {
        constexpr int a = TBL.c[T].a;
        constexpr int b = TBL.c[T].b;
        const float lo = fminf(s[a], s[b]);
        const float hi = fmaxf(s[a], s[b]);
        s[a] = lo;
        s[b] = hi;
        SortNet<T + 1>::run(s);
    }
};
template <>
struct SortNet<NCMP> {
    static __device__ __forceinline__ void run(float (&)[NZ]) {}
};

__device__ __forceinline__ void sort54(float (&s)[NZ]) {
    SortNet<0>::run(s);
}

// arr1[i] = (cumsum_i + gamma)/i, valid iff <= s[i] (i<54) or <= 0 (i==54);
// result = min(valid arr1), fallback arr1[54] if none valid.
// gamma is folded into the running sum's initial value.
__device__ __forceinline__ float spike_mp(const float (&s)[NZ], float gamma) {
    float run  = gamma;
    float best = BIGV;
    float cs_last = 0.0f;
#pragma unroll
    for (int k = 1; k <= NZ; ++k) {
        run += s[k - 1];
        const float cs = run * (1.0f / (float)k);   // compile-time reciprocal
        if (k < NZ) {
            best = (cs <= s[k]) ? fminf(best, cs) : best;
        } else {
            cs_last = cs;
            best = (cs <= 0.0f) ? fminf(best, cs) : best;
        }
    }
    return (best == BIGV) ? cs_last : best;
}

__global__ __launch_bounds__(256) __attribute__((amdgpu_waves_per_eu(2)))
void spike_conv_mp_kernel(const float* __restrict__ x,     // [8,32,32,3]
                          const float* __restrict__ kern,  // [27,64]
                          const int*   __restrict__ gamma_p,
                          float*       __restrict__ out)   // [8,32,32,64]
{
    __shared__ float kerLds[27 * 64];     // 6912 B, row n stride 64
    __shared__ float patchLds[4][32];     // 4 pixels x 27 (padded)

    const int tid = threadIdx.x;
    const int f   = tid & 63;             // filter column
    const int pid = tid >> 6;             // pixel within block (0..3)

    // -------- stage weights: global -> LDS via CDNA5 async DMA path --------
    {
        const uint32_t ldsBase = (uint32_t)(uintptr_t)(&kerLds[0]); // LDS byte offset
        // 27*64*4 = 6912 B = 432 x b128 chunks
        {
            const uint32_t lo = ldsBase + (uint32_t)tid * 16u;
            const uint32_t go = (uint32_t)tid * 16u;
            asm volatile("global_load_async_to_lds_b128 %0, %1, %2"
                         :: "v"(lo), "v"(go), "s"(kern) : "memory");
        }
        if (tid < 432 - 256) {
            const uint32_t lo = ldsBase + (uint32_t)(tid + 256) * 16u;
            const uint32_t go = (uint32_t)(tid + 256) * 16u;
            asm volatile("global_load_async_to_lds_b128 %0, %1, %2"
                         :: "v"(lo), "v"(go), "s"(kern) : "memory");
        }
    }

    // -------- stage 27 patch values per pixel (SAME zero padding) ----------
    const int g  = blockIdx.x * 4 + pid;  // flat pixel id, 0..8191
    const int b  = g >> 10;
    const int hw = g & 1023;
    const int h  = hw >> 5;
    const int w  = hw & 31;

    if (f < 27) {
        const int c  = f % 3;
        const int t  = f / 3;
        const int dj = t % 3;
        const int di = t / 3;
        const int hh = h + di - 1;
        const int ww = w + dj - 1;
        float v = 0.0f;
        if (hh >= 0 && hh < 32 && ww >= 0 && ww < 32)
            v = x[(((b * 32 + hh) * 32) + ww) * 3 + c];
        patchLds[pid][f] = v;
    }

    asm volatile("s_wait_asynccnt 0" ::: "memory");
    __syncthreads();

    const float gamma = (float)(*gamma_p);
    const float* kc = &kerLds[f];           // column f, row stride 64
    const float* pp = &patchLds[pid][0];    // broadcast reads within wave

    float s[NZ];

    // ---------------- z_plus: [3.5+p+0.5k | 3.5-p-0.5k] --------------------
#pragma unroll
    for (int n = 0; n < 27; ++n) {
        const float pn = pp[n];
        const float kk = kc[n * 64];
        s[n]      = fmaf( 0.5f, kk, 3.5f + pn);
        s[27 + n] = fmaf(-0.5f, kk, 3.5f - pn);
    }
    sort54(s);
    const float tp = spike_mp(s, gamma);

    // Liveness fence: force pass 2 to re-read LDS instead of carrying ~54
    // cached values in VGPRs across the whole first sort network.
    asm volatile("" ::: "memory");

    // ---------------- z_minus: [3.5+p-0.5k | 3.5-p+0.5k] -------------------
#pragma unroll
    for (int n = 0; n < 27; ++n) {
        const float pn = pp[n];
        const float kk = kc[n * 64];
        s[n]      = fmaf(-0.5f, kk, 3.5f + pn);
        s[27 + n] = fmaf( 0.5f, kk, 3.5f - pn);
    }
    sort54(s);
    const float tm = spike_mp(s, gamma);

    out[g * 64 + f] = fmaxf(tp - tm, 0.0f);   // coalesced across f
}

extern "C" void kernel_launch(void* const* d_in, const int* in_sizes, int n_in,
                              void* d_out, int out_size, void* d_ws, size_t ws_size,
                              hipStream_t stream) {
    (void)in_sizes; (void)n_in; (void)out_size; (void)d_ws; (void)ws_size;
    const float* x     = (const float*)d_in[0];   // [8,32,32,3] f32
    const float* kern  = (const float*)d_in[1];   // [27,64]     f32
    const int*   gamma = (const int*)d_in[2];     // scalar int
    // d_in[3] = kernel_size (3), baked into the kernel structure.
    float* out = (float*)d_out;                   // [8,32,32,64] f32

    dim3 grid(8 * 32 * 32 / 4);   // 2048 blocks, 4 pixels each
    dim3 block(256);              // 4 pixels x 64 filters, 8 waves (wave32)
    spike_conv_mp_kernel<<<grid, block, 0, stream>>>(x, kern, gamma, out);
}